// Coords2TypedCoords_77610059038770
// MI455X (gfx1250) — compile-verified
//
#include <hip/hip_runtime.h>
#include <hip/hip_bf16.h>

// ---------------------------------------------------------------------------
// Stable counting sort of 4M atoms into 11 type buckets (coords permutation).
// Pure bandwidth problem: ~128MB total traffic -> ~5.5us roofline @ 23.3TB/s.
// CDNA5 features: wave32 ballots (scalar-accumulated popcounts) for stable
// ranks, async global->LDS double-buffering of the type stream (ASYNCcnt via
// inline asm), global_prefetch_b8 for the coordinate stream.
// ---------------------------------------------------------------------------

#define NUM_TYPES   11
#define TYPE_SLOTS  16          // padded type slots in LDS (invalid type = 15)
#define BLK         256         // threads per block (8 wave32)
#define CHUNK       4096        // atoms per block
#define TILES       (CHUNK / BLK)        // 16 (phase 3 tiles)
#define TILES4      (CHUNK / (4 * BLK))  // 4  (phase 1 int4 tiles)
#define MAXB        1024        // max blocks supported by phase2 LDS scan

#define AS3 __attribute__((address_space(3)))

#define USE_ASYNC_LDS 1

__device__ __forceinline__ unsigned ballot32(bool p) {
#if __has_builtin(__builtin_amdgcn_ballot_w32)
    return __builtin_amdgcn_ballot_w32(p);
#else
    return (unsigned)__ballot(p);
#endif
}

#if USE_ASYNC_LDS
// Async copy of one dword from global memory into LDS (ASYNCcnt += 1).
// VDST = VGPR with LDS byte offset; VADDR = 64-bit global address; no SADDR.
__device__ __forceinline__ void async_load_dword_to_lds(const int* gsrc, int* lds_dst) {
    const unsigned ldsOff = (unsigned)(size_t)(AS3 int*)lds_dst;  // generic -> LDS byte offset
    const unsigned long long ga = (unsigned long long)(size_t)gsrc;
    asm volatile("global_load_async_to_lds_b32 %0, %1, off"
                 :: "v"(ldsOff), "v"(ga)
                 : "memory");
}

__device__ __forceinline__ void wait_asynccnt_le1() {
    asm volatile("s_wait_asynccnt 0x1" ::: "memory");
}
#endif

// ---------------------------------------------------------------------------
// Phase 1: per-block histogram. b128 type loads (4 atoms/lane/iter) feed the
// ballot compares; popcounts of the (wave-uniform) ballot masks accumulate in
// SGPRs (s_bcnt/s_add co-issue with VALU). One LDS atomic per wave per type.
// hist layout: [block][NUM_TYPES]
// ---------------------------------------------------------------------------
__global__ void __launch_bounds__(BLK)
phase1_histogram(const int* __restrict__ types, int* __restrict__ hist, int n) {
    __shared__ int h[TYPE_SLOTS];
    const int tid  = threadIdx.x;
    const int b    = blockIdx.x;
    const unsigned lane = tid & 31u;

    if (tid < TYPE_SLOTS) h[tid] = 0;
    __syncthreads();

    const int base = b * CHUNK;

    int wsum[NUM_TYPES];                     // wave-uniform -> SGPRs
    #pragma unroll
    for (int tt = 0; tt < NUM_TYPES; ++tt) wsum[tt] = 0;

    #pragma unroll
    for (int j = 0; j < TILES4; ++j) {
        const int i0 = base + (j * BLK + tid) * 4;   // 16B-aligned
        int4 v;
        if (i0 + 3 < n) {
            v = *(const int4*)(types + i0);          // global_load_b128
        } else {
            v.x = (i0 + 0 < n) ? types[i0 + 0] : 15;
            v.y = (i0 + 1 < n) ? types[i0 + 1] : 15;
            v.z = (i0 + 2 < n) ? types[i0 + 2] : 15;
            v.w = (i0 + 3 < n) ? types[i0 + 3] : 15;
        }
        const int ts[4] = { v.x, v.y, v.z, v.w };
        #pragma unroll
        for (int s = 0; s < 4; ++s) {
            #pragma unroll
            for (int tt = 0; tt < NUM_TYPES; ++tt) {
                wsum[tt] += __popc(ballot32(ts[s] == tt));  // uniform: SGPR math
            }
        }
    }

    // lane tt contributes wsum[tt]: cndmask-select then one atomic per lane<T
    int mine = wsum[0];
    #pragma unroll
    for (int tt = 1; tt < NUM_TYPES; ++tt)
        if (lane == (unsigned)tt) mine = wsum[tt];
    if (lane < NUM_TYPES) atomicAdd(&h[lane], mine);
    __syncthreads();
    if (tid < NUM_TYPES) hist[b * NUM_TYPES + tid] = h[tid];
}

// ---------------------------------------------------------------------------
// Phase 2: single block. Exclusive scan of per-block histograms (type-major),
// type-offset scan, write per-block per-type global start offsets in place,
// and emit counts/offsets (as float, per harness output dtype).
// ---------------------------------------------------------------------------
__global__ void __launch_bounds__(1024)
phase2_scan(const int* __restrict__ hist, int* __restrict__ blockOff,
            float* __restrict__ outCounts, float* __restrict__ outOffsets, int B) {
    __shared__ int h[MAXB * NUM_TYPES];   // transposed: [type][block]
    __shared__ int cnt[NUM_TYPES];
    __shared__ int toff[NUM_TYPES];
    const int tid = threadIdx.x;
    const int total = B * NUM_TYPES;

    for (int idx = tid; idx < total; idx += blockDim.x) {
        const int b = idx / NUM_TYPES, t = idx % NUM_TYPES;
        h[t * B + b] = hist[idx];
    }
    __syncthreads();

    if (tid < NUM_TYPES) {            // serial exclusive scan over blocks, one type per thread
        int run = 0;
        const int rowBase = tid * B;
        for (int b = 0; b < B; ++b) {
            const int v = h[rowBase + b];
            h[rowBase + b] = run;
            run += v;
        }
        cnt[tid] = run;
    }
    __syncthreads();

    if (tid == 0) {                   // tiny exclusive scan over types
        int run = 0;
        for (int t = 0; t < NUM_TYPES; ++t) { toff[t] = run; run += cnt[t]; }
    }
    __syncthreads();

    for (int idx = tid; idx < total; idx += blockDim.x) {
        const int b = idx / NUM_TYPES, t = idx % NUM_TYPES;
        blockOff[idx] = h[t * B + b] + toff[t];   // in-place over hist is safe here
    }
    if (tid < NUM_TYPES) {
        outCounts[tid]  = (float)cnt[tid];
        outOffsets[tid] = (float)toff[tid];
    }
}

// ---------------------------------------------------------------------------
// Phase 3: stable scatter. Types stream through LDS via double-buffered async
// global->LDS loads; stable rank = running cursor + earlier-wave counts +
// intra-wave ballot rank; 12B coord copy to the computed slot.
// ---------------------------------------------------------------------------
__global__ void __launch_bounds__(BLK)
phase3_scatter(const float* __restrict__ coords, const int* __restrict__ types,
               const int* __restrict__ blockOff, float* __restrict__ out, int n) {
    __shared__ int tbuf[2][BLK];              // double-buffered types
    __shared__ int wcnt[BLK / 32][TYPE_SLOTS];
    __shared__ int runOff[TYPE_SLOTS];

    const int tid = threadIdx.x;
    const int b   = blockIdx.x;
    const unsigned lane  = tid & 31u;
    const unsigned wav   = (unsigned)tid >> 5;
    const unsigned below = (1u << lane) - 1u;
    const int base = b * CHUNK;

    if (tid < NUM_TYPES) runOff[tid] = blockOff[b * NUM_TYPES + tid];

#if USE_ASYNC_LDS
    {   // prologue: fill both buffers (tiles 0 and 1); clamp addresses in-range
        int i0 = base + tid;           if (i0 >= n) i0 = n - 1;
        int i1 = base + BLK + tid;     if (i1 >= n) i1 = n - 1;
        async_load_dword_to_lds(types + i0, &tbuf[0][tid]);
        async_load_dword_to_lds(types + i1, &tbuf[1][tid]);
    }
#endif
    __syncthreads();   // runOff ready

    for (int it = 0; it < TILES; ++it) {
        const int i = base + it * BLK + tid;
        int t;
#if USE_ASYNC_LDS
        wait_asynccnt_le1();                 // tile `it` resident in LDS
        t = tbuf[it & 1][tid];
#else
        { int ic = i < n ? i : (n - 1); t = types[ic]; }
#endif
        if (i >= n) t = 15;                  // invalid lanes -> padded slot, never stored

        // prefetch next tile's coordinates into cache (global_prefetch_b8)
        {
            int ipf = i + BLK; if (ipf >= n) ipf = n - 1;
            __builtin_prefetch(&coords[3 * (size_t)ipf], 0, 0);
        }

        // stable intra-wave rank + per-wave per-type counts via wave32 ballots
        int intra = 0;
        int mycnt = 0;                       // lane tt keeps count of type tt
        #pragma unroll
        for (int tt = 0; tt < NUM_TYPES; ++tt) {
            const unsigned m = ballot32(t == tt);
            if (t == tt) intra = __popc(m & below);
            if (lane == (unsigned)tt) mycnt = __popc(m);
        }
        if (lane < NUM_TYPES) wcnt[wav][lane] = mycnt;   // one ds_store per wave
        __syncthreads();                     // wcnt visible

        int dst = 0;
        if (t < NUM_TYPES) {
            dst = runOff[t] + intra;
            for (unsigned w = 0; w < wav; ++w) dst += wcnt[w][t];
        }

        if (i < n) {                         // 12B move; same-type dsts are contiguous
            const size_t s = 3 * (size_t)i;
            const size_t d = 3 * (size_t)dst;
            const float x = coords[s + 0];
            const float y = coords[s + 1];
            const float z = coords[s + 2];
            out[d + 0] = x; out[d + 1] = y; out[d + 2] = z;
        }
        __syncthreads();                     // all runOff/wcnt reads done

        if (tid < NUM_TYPES) {               // advance running cursor by tile totals
            int tot = 0;
            #pragma unroll
            for (int w = 0; w < BLK / 32; ++w) tot += wcnt[w][tid];
            runOff[tid] += tot;
        }

#if USE_ASYNC_LDS
        {   // issue tile it+2 into the buffer just consumed (own lanes already read it)
            int inx = base + (it + 2) * BLK + tid; if (inx >= n) inx = n - 1;
            async_load_dword_to_lds(types + inx, &tbuf[it & 1][tid]);
        }
#endif
        __syncthreads();                     // runOff update visible; wcnt reusable
    }
}

// ---------------------------------------------------------------------------
extern "C" void kernel_launch(void* const* d_in, const int* in_sizes, int n_in,
                              void* d_out, int out_size, void* d_ws, size_t ws_size,
                              hipStream_t stream) {
    const float* coords = (const float*)d_in[0];
    const int*   types  = (const int*)d_in[1];
    // d_in[2] = num_atom_types (scalar, == NUM_TYPES); compile-time constant used.

    const int n = in_sizes[1];                      // number of atoms
    const int B = (n + CHUNK - 1) / CHUNK;          // 977 for n = 4,000,000 (<= MAXB)

    int*   hist       = (int*)d_ws;                 // [B][NUM_TYPES], scanned in place
    float* out        = (float*)d_out;              // typed coords [3n]
    float* outCounts  = out + (size_t)3 * n;        // [NUM_TYPES]
    float* outOffsets = outCounts + NUM_TYPES;      // [NUM_TYPES]

    phase1_histogram<<<B, BLK, 0, stream>>>(types, hist, n);
    phase2_scan<<<1, 1024, 0, stream>>>(hist, hist, outCounts, outOffsets, B);
    phase3_scatter<<<B, BLK, 0, stream>>>(coords, types, hist, out, n);
}